// SGC_16604343566786
// MI455X (gfx1250) — compile-verified
//
#include <hip/hip_runtime.h>

typedef float v2f __attribute__((ext_vector_type(2)));
typedef float v8f __attribute__((ext_vector_type(8)));

#define THREADS 256

// ---------------- degree / normalization ----------------

__global__ void sgc_init_deg(float* deg, int N) {
  int i = blockIdx.x * THREADS + threadIdx.x;
  if (i < N) deg[i] = 1.0f;  // self-loop contribution
}

__global__ void sgc_deg(const long long* __restrict__ dst, float* deg, int E) {
  int e = blockIdx.x * THREADS + threadIdx.x;
  if (e < E) unsafeAtomicAdd(&deg[dst[e]], 1.0f);
}

__global__ void sgc_rsqrt(float* deg, int N) {
  int i = blockIdx.x * THREADS + threadIdx.x;
  if (i < N) {
    float d = deg[i];
    deg[i] = d > 0.0f ? rsqrtf(d) : 0.0f;
  }
}

// ---------------- propagation ----------------

// h_out = h_in * dis^2  (self-loop edge; also fully initializes h_out)
__global__ void sgc_selfloop(const float* __restrict__ h_in, const float* __restrict__ dis,
                             float* __restrict__ h_out, long totalV4 /* N*32 float4s */) {
  long idx = (long)blockIdx.x * THREADS + threadIdx.x;
  if (idx >= totalV4) return;
  long node = idx >> 5;           // 32 float4 per node (128 ch)
  float s = dis[node];
  s *= s;
  float4 v = ((const float4*)h_in)[idx];
  float4 r = make_float4(v.x * s, v.y * s, v.z * s, v.w * s);
  ((float4*)h_out)[idx] = r;
}

// one wave32 per edge: gather float4/lane, scale, 4 hw fp32 atomics/lane
__global__ void sgc_edges(const long long* __restrict__ src, const long long* __restrict__ dst,
                          const float* __restrict__ dis, const float* __restrict__ h_in,
                          float* __restrict__ h_out, int E) {
  int e = blockIdx.x * (THREADS / 32) + (threadIdx.x >> 5);
  if (e >= E) return;
  int lane = threadIdx.x & 31;
  long s = (long)src[e];
  long d = (long)dst[e];
  float norm = dis[s] * dis[d];
  float4 v = ((const float4*)(h_in + s * 128))[lane];
  float* o = h_out + d * 128 + (long)lane * 4;
  unsafeAtomicAdd(o + 0, v.x * norm);
  unsafeAtomicAdd(o + 1, v.y * norm);
  unsafeAtomicAdd(o + 2, v.z * norm);
  unsafeAtomicAdd(o + 3, v.w * norm);
}

// ---------------- final projection via V_WMMA_F32_16X16X4_F32 ----------------
// out[N,64] = H[N,128] @ W[64,128]^T + b
// One wave per 16x16 output tile; K=128 in 32 chained 16x16x4 fp32 WMMAs.
__global__ void sgc_gemm_wmma(const float* __restrict__ H, const float* __restrict__ Wt,
                              const float* __restrict__ bias, float* __restrict__ out,
                              int nTilesM) {
  int wave = threadIdx.x >> 5;
  int lane = threadIdx.x & 31;
  int tile = blockIdx.x * (THREADS / 32) + wave;
  int total = nTilesM * 4;                 // 64/16 = 4 col tiles
  if (tile >= total) return;               // wave-uniform branch: EXEC stays all-1s
  int mTile = tile >> 2;
  int oTile = tile & 3;
  int m0 = mTile * 16, o0 = oTile * 16;
  int half = lane >> 4;                    // 0: K+{0,1}, 1: K+{2,3}
  int l = lane & 15;

  const float* aRow = H  + (size_t)(m0 + l) * 128 + 2 * half;  // A[m][k]
  const float* bRow = Wt + (size_t)(o0 + l) * 128 + 2 * half;  // B[k][n] = W[n][k]

  v8f acc = {};
#pragma unroll
  for (int kk = 0; kk < 128; kk += 4) {
    v2f a = { aRow[kk], aRow[kk + 1] };
    v2f b = { bRow[kk], bRow[kk + 1] };
    acc = __builtin_amdgcn_wmma_f32_16x16x4_f32(
        /*neg_a=*/false, a, /*neg_b=*/false, b,
        /*c_mod=*/(short)0, acc, /*reuse_a=*/false, /*reuse_b=*/false);
  }

  float bv = bias[o0 + l];
#pragma unroll
  for (int r = 0; r < 8; ++r) {
    // D layout: VGPR r -> row m0+r (lanes 0-15) / m0+r+8 (lanes 16-31), col o0+l
    out[(size_t)(m0 + r + 8 * half) * 64 + o0 + l] = acc[r] + bv;
  }
}

// scalar tail for rows not covered by 16-row tiles (N=100000 -> unused, kept for safety)
__global__ void sgc_gemm_tail(const float* __restrict__ H, const float* __restrict__ Wt,
                              const float* __restrict__ bias, float* __restrict__ out,
                              int N, int start) {
  int row = start + blockIdx.x;
  if (row >= N) return;
  int o = threadIdx.x;  // 64 threads
  float s = bias[o];
  const float* h = H + (size_t)row * 128;
  const float* w = Wt + (size_t)o * 128;
  for (int k = 0; k < 128; ++k) s += h[k] * w[k];
  out[(size_t)row * 64 + o] = s;
}

// ---------------- driver ----------------

extern "C" void kernel_launch(void* const* d_in, const int* in_sizes, int n_in,
                              void* d_out, int out_size, void* d_ws, size_t ws_size,
                              hipStream_t stream) {
  const float*     x  = (const float*)d_in[0];
  const long long* ei = (const long long*)d_in[1];  // int64 edge_index [2,E]
  const float*     W  = (const float*)d_in[2];      // [64,128]
  const float*     b  = (const float*)d_in[3];      // [64]
  const int N = in_sizes[0] / 128;
  const int E = in_sizes[1] / 2;
  const int HOPS_ = 3;  // constant in reference

  const long long* srcI = ei;       // edge_index[0] = message sources
  const long long* dstI = ei + E;   // edge_index[1] = aggregation targets

  char* ws = (char*)d_ws;
  size_t off = 0;
  float* dis  = (float*)(ws + off); off += ((size_t)N * 4 + 255) & ~(size_t)255;
  float* bufA = (float*)(ws + off); off += (size_t)N * 128 * 4;
  float* bufB = (float*)(ws + off);

  dim3 blk(THREADS);
  sgc_init_deg<<<(N + THREADS - 1) / THREADS, blk, 0, stream>>>(dis, N);
  sgc_deg<<<(E + THREADS - 1) / THREADS, blk, 0, stream>>>(dstI, dis, E);
  sgc_rsqrt<<<(N + THREADS - 1) / THREADS, blk, 0, stream>>>(dis, N);

  long totalV4 = (long)N * 32;
  int vBlocks = (int)((totalV4 + THREADS - 1) / THREADS);
  int eBlocks = (E + (THREADS / 32) - 1) / (THREADS / 32);

  const float* hin = x;
  float* hout = bufA;
  for (int hop = 0; hop < HOPS_; ++hop) {
    sgc_selfloop<<<vBlocks, blk, 0, stream>>>(hin, dis, hout, totalV4);
    sgc_edges<<<eBlocks, blk, 0, stream>>>(srcI, dstI, dis, hin, hout, E);
    hin = hout;
    hout = (hout == bufA) ? bufB : bufA;
  }

  int nTilesM = N / 16;
  if (nTilesM > 0) {
    int totalTiles = nTilesM * 4;
    int gBlocks = (totalTiles + (THREADS / 32) - 1) / (THREADS / 32);
    sgc_gemm_wmma<<<gBlocks, blk, 0, stream>>>(hin, W, b, (float*)d_out, nTilesM);
  }
  int rem = N & 15;
  if (rem) {
    sgc_gemm_tail<<<rem, 64, 0, stream>>>(hin, W, b, (float*)d_out, N, N & ~15);
  }
}